// IGNN_23141283791621
// MI455X (gfx1250) — compile-verified
//
#include <hip/hip_runtime.h>
#include <cmath>

// ---------------------------------------------------------------------------
// IGNN forward for MI455X (gfx1250, wave32, WMMA).
//
// * Activations are node-major [n, m] so SpMM touches contiguous m-float rows
//   and WMMA A-tiles (16 nodes x 4 k) are row-contiguous.
// * Dense GEMMs use V_WMMA_F32_16X16X4_F32 (full fp32; the problem is
//   L2-bound at ~21 GB of sparse traffic vs ~33 GFLOP dense, so dropping to
//   bf16 buys nothing).
// * SpMM (X @ A) is the bottleneck: instead of float-atomic scatter we build
//   a column-grouped CSR of A once per call, then each SpMM is an atomic-free
//   gather with the "+B" and relu fused into its epilogue.
// ---------------------------------------------------------------------------

#define KAPPA_F 0.9f

typedef __attribute__((ext_vector_type(2))) float v2f;
typedef __attribute__((ext_vector_type(8))) float v8f;

// ======================= elementwise helpers ===============================

__global__ void fill_f32(float* __restrict__ p, float v, long count) {
  long i = (long)blockIdx.x * blockDim.x + threadIdx.x;
  if (i < count) p[i] = v;
}

// transpose [rows, cols] -> [cols, rows]   (features [p,n] -> x [n,p])
__global__ void transpose_f32(const float* __restrict__ in,
                              float* __restrict__ out, int rows, int cols) {
  __shared__ float tile[32][33];
  int c0 = blockIdx.x * 32, r0 = blockIdx.y * 32;
  for (int dy = 0; dy < 32; dy += 8) {
    int r = r0 + threadIdx.y + dy, c = c0 + threadIdx.x;
    if (r < rows && c < cols)
      tile[threadIdx.y + dy][threadIdx.x] = in[(long)r * cols + c];
  }
  __syncthreads();
  for (int dy = 0; dy < 32; dy += 8) {
    int oc = r0 + threadIdx.x;         // out col = in row
    int orow = c0 + threadIdx.y + dy;  // out row = in col
    if (orow < cols && oc < rows)
      out[(long)orow * rows + oc] = tile[threadIdx.x][threadIdx.y + dy];
  }
}

// ======================= power iteration ===================================

// w[rows[e]] += vals[e] * v[cols[e]]
__global__ void spmv_scatter(const int* __restrict__ rows,
                             const int* __restrict__ cols,
                             const float* __restrict__ vals,
                             const float* __restrict__ v,
                             float* __restrict__ w, int E) {
  int e = blockIdx.x * blockDim.x + threadIdx.x;
  if (e < E) unsafeAtomicAdd(&w[rows[e]], vals[e] * v[cols[e]]);
}

__global__ void sumsq_f32(const float* __restrict__ w, float* __restrict__ s,
                          int n) {
  __shared__ float red[256];
  float acc = 0.f;
  for (long i = (long)blockIdx.x * blockDim.x + threadIdx.x; i < n;
       i += (long)gridDim.x * blockDim.x)
    acc += w[i] * w[i];
  red[threadIdx.x] = acc;
  __syncthreads();
  for (int off = 128; off > 0; off >>= 1) {
    if ((int)threadIdx.x < off) red[threadIdx.x] += red[threadIdx.x + off];
    __syncthreads();
  }
  if (threadIdx.x == 0) unsafeAtomicAdd(s, red[0]);
}

// v = w / (||w|| + 1e-12);  a_rho = ||w||  (last iteration's norm survives)
__global__ void pow_scale(const float* __restrict__ s,
                          const float* __restrict__ w, float* __restrict__ v,
                          float* __restrict__ a_rho, int n) {
  float nrm = sqrtf(*s);
  long i = (long)blockIdx.x * blockDim.x + threadIdx.x;
  if (i < n) v[i] = w[i] / (nrm + 1e-12f);
  if (i == 0) *a_rho = nrm;
}

// ======================= CSR build (group edges by column) =================

__global__ void hist_cols(const int* __restrict__ cols, int* __restrict__ cnt,
                          int E) {
  int e = blockIdx.x * blockDim.x + threadIdx.x;
  if (e < E) atomicAdd(&cnt[cols[e]], 1);
}

// per-block inclusive scan, emits block totals
__global__ void scan_block(const int* __restrict__ in, int* __restrict__ out,
                           int* __restrict__ bsum, int n) {
  __shared__ int sh[256];
  int i = blockIdx.x * 256 + threadIdx.x;
  int v = (i < n) ? in[i] : 0;
  sh[threadIdx.x] = v;
  __syncthreads();
  for (int off = 1; off < 256; off <<= 1) {
    int add = ((int)threadIdx.x >= off) ? sh[threadIdx.x - off] : 0;
    __syncthreads();
    sh[threadIdx.x] += add;
    __syncthreads();
  }
  if (i < n) out[i] = sh[threadIdx.x];
  if (threadIdx.x == 255) bsum[blockIdx.x] = sh[255];
}

// single-block inclusive scan of block totals (B <= 1024)
__global__ void scan_sums(int* __restrict__ bsum, int B) {
  __shared__ int sh[1024];
  int t = threadIdx.x;
  sh[t] = (t < B) ? bsum[t] : 0;
  __syncthreads();
  for (int off = 1; off < 1024; off <<= 1) {
    int add = (t >= off) ? sh[t - off] : 0;
    __syncthreads();
    sh[t] += add;
    __syncthreads();
  }
  if (t < B) bsum[t] = sh[t];
}

__global__ void scan_add(int* __restrict__ out, const int* __restrict__ bsum,
                         int n) {
  int i = blockIdx.x * 256 + threadIdx.x;
  if (i < n && blockIdx.x > 0) out[i] += bsum[blockIdx.x - 1];
}

__global__ void cursor_init(const int* __restrict__ inc,
                            const int* __restrict__ cnt, int* __restrict__ cur,
                            int n) {
  int i = blockIdx.x * blockDim.x + threadIdx.x;
  if (i < n) cur[i] = inc[i] - cnt[i];
}

// permute (row, val) into column-grouped order
__global__ void csr_fill(const int* __restrict__ rows,
                         const int* __restrict__ cols,
                         const float* __restrict__ vals, int* __restrict__ cur,
                         int* __restrict__ srcRow, float* __restrict__ sval,
                         int E) {
  int e = blockIdx.x * blockDim.x + threadIdx.x;
  if (e < E) {
    int pos = atomicAdd(&cur[cols[e]], 1);
    srcRow[pos] = rows[e];
    sval[pos] = vals[e];
  }
}

// ======================= SpMM gather (Y = [relu]( G @ A [+ B] )) ===========
// One thread per (output column c, feature i). Edge segment of column c is
// [inc[c]-cnt[c], inc[c]); per-edge data is wave-uniform, X reads contiguous.
template <bool ADDB, bool RELU>
__global__ void spmm_gather_k(const int* __restrict__ inc,
                              const int* __restrict__ cnt,
                              const int* __restrict__ srcRow,
                              const float* __restrict__ sval,
                              const float* __restrict__ G,
                              const float* __restrict__ Bm,
                              float* __restrict__ Y, int n, int mShift) {
  long t = (long)blockIdx.x * blockDim.x + threadIdx.x;
  int c = (int)(t >> mShift);
  if (c >= n) return;
  int i = (int)t & ((1 << mShift) - 1);
  int end = inc[c];
  int start = end - cnt[c];
  __builtin_prefetch(&sval[start], 0, 1);    // global_prefetch_b8
  __builtin_prefetch(&srcRow[start], 0, 1);
  float acc = ADDB ? Bm[((long)c << mShift) + i] : 0.f;
  for (int j = start; j < end; ++j)
    acc = fmaf(sval[j], G[((long)srcRow[j] << mShift) + i], acc);
  if (RELU) acc = fmaxf(acc, 0.f);
  Y[((long)c << mShift) + i] = acc;
}

// ======================= proj_linf (row simplex projection) ================
template <int MDIM>
__global__ void proj_linf_k(const float* __restrict__ W,
                            float* __restrict__ Wp,
                            const float* __restrict__ a_rho) {
  __shared__ float u[MDIM];
  __shared__ float cs[MDIM];
  __shared__ int ired[MDIM];
  __shared__ float ssum;
  __shared__ int srho;

  int row = blockIdx.x, t = threadIdx.x;
  float wv = W[row * MDIM + t];
  float aw = fabsf(wv);
  float vv = KAPPA_F / (*a_rho);

  cs[t] = aw;
  __syncthreads();
  for (int off = MDIM / 2; off > 0; off >>= 1) {
    if (t < off) cs[t] += cs[t + off];
    __syncthreads();
  }
  if (t == 0) ssum = cs[0];
  u[t] = aw;
  __syncthreads();
  float s = ssum;

  // bitonic sort descending
  for (int ksz = 2; ksz <= MDIM; ksz <<= 1) {
    for (int j = ksz >> 1; j > 0; j >>= 1) {
      int ixj = t ^ j;
      if (ixj > t) {
        float x0 = u[t], x1 = u[ixj];
        bool desc = ((t & ksz) == 0);
        bool sw = desc ? (x0 < x1) : (x0 > x1);
        if (sw) { u[t] = x1; u[ixj] = x0; }
      }
      __syncthreads();
    }
  }

  cs[t] = u[t];
  __syncthreads();
  for (int off = 1; off < MDIM; off <<= 1) {
    float add = (t >= off) ? cs[t - off] : 0.f;
    __syncthreads();
    cs[t] += add;
    __syncthreads();
  }

  float cssv = cs[t] - vv;
  ired[t] = (u[t] * (float)(t + 1) > cssv) ? 1 : 0;
  __syncthreads();
  for (int off = MDIM / 2; off > 0; off >>= 1) {
    if (t < off) ired[t] += ired[t + off];
    __syncthreads();
  }
  if (t == 0) srho = ired[0];
  __syncthreads();

  int rho = srho < 1 ? 1 : srho;
  float theta = (cs[rho - 1] - vv) / (float)rho;
  float projv = copysignf(fmaxf(aw - theta, 0.f), wv);
  Wp[row * MDIM + t] = (s > vv) ? projv : wv;
}

// ======================= WMMA fp32 GEMM ====================================
// Out[n, M] = act( Xin[n, K] @ Wt[M, K]^T  (+ Cin[n, M] + bias[M]) )
template <int MT, bool HASC, bool EACT>
__global__ __launch_bounds__(256) void gemm_wmma_k(
    const float* __restrict__ Xin, const float* __restrict__ Wt,
    const float* __restrict__ Cin, const float* __restrict__ bias,
    float* __restrict__ Out, int n, int K) {
  constexpr int M = MT * 16;
  extern __shared__ float ldsW[];  // M*K floats

  for (int idx = threadIdx.x; idx < M * K; idx += blockDim.x)
    ldsW[idx] = Wt[idx];
  __syncthreads();

  const int lane = threadIdx.x & 31;
  const int wave = threadIdx.x >> 5;
  const int tile = blockIdx.x * 8 + wave;   // uniform per wave
  if (tile >= (n >> 4)) return;             // whole-wave exit
  const int node0 = tile << 4;
  const int nIdx = lane & 15;
  const int khalf = lane >> 4;
  const int rowOff = khalf * 8;

  v8f acc[MT];
#pragma unroll
  for (int mt = 0; mt < MT; ++mt) {
    if (HASC) {
      float b = bias[mt * 16 + nIdx];
#pragma unroll
      for (int i = 0; i < 8; ++i)
        acc[mt][i] =
            Cin[(long)(node0 + rowOff + i) * M + mt * 16 + nIdx] + b;
    } else {
#pragma unroll
      for (int i = 0; i < 8; ++i) acc[mt][i] = 0.f;
    }
  }

  const float* aBase = Xin + (long)(node0 + nIdx) * K + khalf * 2;
  for (int k0 = 0; k0 < K; k0 += 4) {
    v2f a;
    a.x = aBase[k0];
    a.y = aBase[k0 + 1];
#pragma unroll
    for (int mt = 0; mt < MT; ++mt) {
      const float* bp = &ldsW[(mt * 16 + nIdx) * K + k0 + khalf * 2];
      v2f b;
      b.x = bp[0];
      b.y = bp[1];
      acc[mt] = __builtin_amdgcn_wmma_f32_16x16x4_f32(
          false, a, false, b, (short)0, acc[mt], false, false);
    }
  }

#pragma unroll
  for (int mt = 0; mt < MT; ++mt) {
#pragma unroll
    for (int i = 0; i < 8; ++i) {
      float v = acc[mt][i];
      if (EACT) v = (v > 0.f) ? v : (__expf(v) - 1.f);
      Out[(long)(node0 + rowOff + i) * M + mt * 16 + nIdx] = v;
    }
  }
}

// ======================= host-side dispatch ================================

static void gemm_launch(int M, int K, const float* Xin, const float* Wt,
                        const float* Cin, const float* bias, float* Out,
                        int n, bool elu, hipStream_t s) {
  dim3 g((unsigned)((n / 16 + 7) / 8)), b(256);
  size_t lds = (size_t)M * K * sizeof(float);
#define GL(MT)                                                                \
  do {                                                                        \
    if (Cin) {                                                                \
      if (elu)                                                                \
        gemm_wmma_k<MT, true, true><<<g, b, lds, s>>>(Xin, Wt, Cin, bias,     \
                                                      Out, n, K);             \
      else                                                                    \
        gemm_wmma_k<MT, true, false><<<g, b, lds, s>>>(Xin, Wt, Cin, bias,    \
                                                       Out, n, K);            \
    } else                                                                    \
      gemm_wmma_k<MT, false, false><<<g, b, lds, s>>>(Xin, Wt, Cin, bias,     \
                                                      Out, n, K);             \
  } while (0)
  switch (M / 16) {
    case 1: GL(1); break;
    case 2: GL(2); break;
    case 4: GL(4); break;
    case 8: GL(8); break;
    default: break;
  }
#undef GL
}

static void gather_launch(int n, int M, const int* inc, const int* cnt,
                          const int* srcRow, const float* sval, const float* G,
                          const float* Bm, float* Y, bool addB, bool relu,
                          hipStream_t s) {
  int sh = 31 - __builtin_clz((unsigned)M);
  long total = (long)n << sh;
  unsigned blocks = (unsigned)((total + 255) / 256);
  if (addB) {
    if (relu)
      spmm_gather_k<true, true><<<blocks, 256, 0, s>>>(inc, cnt, srcRow, sval,
                                                       G, Bm, Y, n, sh);
    else
      spmm_gather_k<true, false><<<blocks, 256, 0, s>>>(inc, cnt, srcRow, sval,
                                                        G, Bm, Y, n, sh);
  } else {
    spmm_gather_k<false, false><<<blocks, 256, 0, s>>>(inc, cnt, srcRow, sval,
                                                       G, Bm, Y, n, sh);
  }
}

static void proj_launch(int M, const float* W, float* Wp, const float* aRho,
                        hipStream_t s) {
  switch (M) {
    case 16: proj_linf_k<16><<<16, 16, 0, s>>>(W, Wp, aRho); break;
    case 32: proj_linf_k<32><<<32, 32, 0, s>>>(W, Wp, aRho); break;
    case 64: proj_linf_k<64><<<64, 64, 0, s>>>(W, Wp, aRho); break;
    case 128: proj_linf_k<128><<<128, 128, 0, s>>>(W, Wp, aRho); break;
    default: break;
  }
}

static inline long up256(long x) { return (x + 255) & ~255L; }

extern "C" void kernel_launch(void* const* d_in, const int* in_sizes, int n_in,
                              void* d_out, int out_size, void* d_ws,
                              size_t ws_size, hipStream_t stream) {
  (void)n_in; (void)out_size; (void)ws_size;

  const float* features = (const float*)d_in[0];
  const int* adj_rows = (const int*)d_in[1];
  const int* adj_cols = (const int*)d_in[2];
  const float* adj_vals = (const float*)d_in[3];

  const float* Wl[5];
  const float* Oml[5];
  const float* Lwl[5];
  const float* Lbl[5];
  int m[5], kom[5];
  for (int i = 0; i < 5; ++i) {
    Wl[i] = (const float*)d_in[4 + 2 * i];
    Oml[i] = (const float*)d_in[5 + 2 * i];
    Lwl[i] = (const float*)d_in[14 + 2 * i];
    Lbl[i] = (const float*)d_in[15 + 2 * i];
    m[i] = (int)(sqrt((double)in_sizes[4 + 2 * i]) + 0.5);
    kom[i] = in_sizes[5 + 2 * i] / m[i];
  }
  const int p = kom[0];
  const int n = in_sizes[0] / p;
  const int E = in_sizes[1];

  // ---- workspace carve-out (~136 MB) ----
  char* ws = (char*)d_ws;
  long off = 0;
  float* scal = (float*)(ws + off); off = up256(off + 64 * 4);  // [0]=s [1]=rho
  float* Wp   = (float*)(ws + off); off = up256(off + 128L * 128 * 4);
  float* vvec = (float*)(ws + off); off = up256(off + (long)n * 4);
  float* wvec = (float*)(ws + off); off = up256(off + (long)n * 4);
  int* cntC   = (int*)(ws + off);   off = up256(off + (long)n * 4);
  int* incC   = (int*)(ws + off);   off = up256(off + (long)n * 4);
  int* curC   = (int*)(ws + off);   off = up256(off + (long)n * 4);
  int* bsum   = (int*)(ws + off);   off = up256(off + 1024L * 4);
  int* srcRow = (int*)(ws + off);   off = up256(off + (long)E * 4);
  float* sval = (float*)(ws + off); off = up256(off + (long)E * 4);
  const long matB = up256((long)n * 128 * 4);
  float* bufA = (float*)(ws + off); off += matB;   // x_cur
  float* bufY = (float*)(ws + off); off += matB;   // x_next
  float* Bm   = (float*)(ws + off); off += matB;   // B
  float* Xm   = (float*)(ws + off); off += matB;   // X (fixed-point state)
  float* Gm   = (float*)(ws + off); off += matB;   // T / G (pre-SpMM)

  // ---- features [p, n] -> x_cur [n, p] ----
  {
    dim3 g((unsigned)((n + 31) / 32), (unsigned)((p + 31) / 32)), b(32, 8);
    transpose_f32<<<g, b, 0, stream>>>(features, bufA, p, n);
  }

  // ---- column-grouped CSR of A (built once, reused by all 55 SpMMs) ----
  {
    int nb = (n + 255) / 256;  // <= 1024 assumed (n <= 262144)
    fill_f32<<<nb, 256, 0, stream>>>((float*)cntC, 0.f, n);  // zero ints
    hist_cols<<<(E + 255) / 256, 256, 0, stream>>>(adj_cols, cntC, E);
    scan_block<<<nb, 256, 0, stream>>>(cntC, incC, bsum, n);
    scan_sums<<<1, 1024, 0, stream>>>(bsum, nb);
    scan_add<<<nb, 256, 0, stream>>>(incC, bsum, n);
    cursor_init<<<nb, 256, 0, stream>>>(incC, cntC, curC, n);
    csr_fill<<<(E + 255) / 256, 256, 0, stream>>>(adj_rows, adj_cols, adj_vals,
                                                  curC, srcRow, sval, E);
  }

  // ---- spectral radius: 50 power iterations (atomic SpMV; tiny) ----
  fill_f32<<<(n + 255) / 256, 256, 0, stream>>>(vvec, 1.f / sqrtf((float)n),
                                                n);
  for (int it = 0; it < 50; ++it) {
    fill_f32<<<(n + 255) / 256, 256, 0, stream>>>(wvec, 0.f, n);
    spmv_scatter<<<(E + 255) / 256, 256, 0, stream>>>(adj_rows, adj_cols,
                                                      adj_vals, vvec, wvec, E);
    fill_f32<<<1, 1, 0, stream>>>(scal, 0.f, 1);
    sumsq_f32<<<512, 256, 0, stream>>>(wvec, scal, n);
    pow_scale<<<(n + 255) / 256, 256, 0, stream>>>(scal, wvec, vvec, scal + 1,
                                                   n);
  }
  const float* aRho = scal + 1;

  // ---- 5 implicit layers ----
  float* xcur = bufA;
  float* ybuf = bufY;
  for (int L = 0; L < 5; ++L) {
    const int M = m[L];
    const int K = kom[L];

    // Wp = proj_linf(W, kappa / a_rho)
    proj_launch(M, Wl[L], Wp, aRho, stream);

    // T = Om @ x -> Gm ; B = T @ A (atomic-free gather)
    gemm_launch(M, K, xcur, Oml[L], nullptr, nullptr, Gm, n, false, stream);
    gather_launch(n, M, incC, cntC, srcRow, sval, Gm, nullptr, Bm,
                  /*addB=*/false, /*relu=*/false, stream);

    // fixed point: X = relu((Wp @ X) @ A + B), X0 = B (read B directly)
    for (int itr = 0; itr < 10; ++itr) {
      const float* xin = (itr == 0) ? Bm : Xm;
      gemm_launch(M, M, xin, Wp, nullptr, nullptr, Gm, n, false, stream);
      gather_launch(n, M, incC, cntC, srcRow, sval, Gm, Bm, Xm,
                    /*addB=*/true, /*relu=*/true, stream);
    }

    if (L < 4) {
      // x_next = elu(ig + Lw @ x + Lb)  (fused via WMMA C-operand)
      gemm_launch(M, K, xcur, Lwl[L], Xm, Lbl[L], ybuf, n, true, stream);
      float* t = xcur; xcur = ybuf; ybuf = t;
    } else {
      // out[n,16] = ig + Vw @ x + Vb  (node-major == reference row-major)
      gemm_launch(M, K, xcur, Lwl[L], Xm, Lbl[L], (float*)d_out, n, false,
                  stream);
    }
  }
}